// JK_70411693850861
// MI455X (gfx1250) — compile-verified
//
#include <hip/hip_runtime.h>

#define N_NODES 100000
#define N_EDGES 1600000
#define NFEAT 128
#define NHID 128
#define NCLASS 40

typedef __attribute__((ext_vector_type(2))) float v2f;
typedef __attribute__((ext_vector_type(8))) float v8f;

// ---------------------------------------------------------------- zero fill
__global__ void gin_zero_kernel(float* __restrict__ p, int n4) {
    int i = blockIdx.x * blockDim.x + threadIdx.x;
    if (i < n4) ((float4*)p)[i] = make_float4(0.f, 0.f, 0.f, 0.f);
}

// ------------------------------------------------- edge scatter: agg[dst] += w * h[src]
// one wave32 per edge; each lane handles 4 consecutive features (32*4 = 128)
__global__ __launch_bounds__(256)
void gin_scatter_kernel(const float* __restrict__ h,
                        const int*   __restrict__ src,
                        const int*   __restrict__ dst,
                        const float* __restrict__ ew,
                        float* __restrict__ agg) {
    int e = blockIdx.x * (blockDim.x >> 5) + (threadIdx.x >> 5);
    if (e >= N_EDGES) return;
    int lane = threadIdx.x & 31;
    int s = src[e];
    int d = dst[e];
    float w = ew[e];
    float4 v = ((const float4*)(h + (size_t)s * NHID))[lane];
    float* ad = agg + (size_t)d * NHID + lane * 4;
    unsafeAtomicAdd(ad + 0, v.x * w);
    unsafeAtomicAdd(ad + 1, v.y * w);
    unsafeAtomicAdd(ad + 2, v.z * w);
    unsafeAtomicAdd(ad + 3, v.w * w);
}

// ---------------------------------------- hout = relu((hin + agg) @ W + b)
// 8 waves per block; block owns 16 rows, wave w owns columns [16w, 16w+16)
// fp32 WMMA 16x16x4: A lane layout (m = lane&15, K = kk + 2*(lane>>4) + {0,1}),
// B lane layout (n = lane&15, same K split), C/D: VGPR v -> M = v + 8*(lane>>4)
__global__ __launch_bounds__(256)
void gin_gemm_relu_kernel(const float* __restrict__ hin,
                          const float* __restrict__ agg,
                          const float* __restrict__ W,
                          const float* __restrict__ bias,
                          float* __restrict__ hout) {
    __shared__ float As[16][NFEAT + 4];   // stride 132 -> conflict-free
    int tile = blockIdx.x;
    int tid  = threadIdx.x;

    for (int i = tid; i < 16 * NFEAT; i += 256) {
        int r = i >> 7;
        int c = i & (NFEAT - 1);
        size_t idx = (size_t)(tile * 16 + r) * NFEAT + c;
        As[r][c] = hin[idx] + agg[idx];
    }
    __syncthreads();

    int wave = tid >> 5;
    int lane = tid & 31;
    int half = lane >> 4;
    int mn   = lane & 15;
    int n0   = wave * 16;

    v8f acc = {0.f, 0.f, 0.f, 0.f, 0.f, 0.f, 0.f, 0.f};
    for (int kk = 0; kk < NFEAT; kk += 4) {
        int k0 = kk + 2 * half;
        v2f a, b;
        a[0] = As[mn][k0];
        a[1] = As[mn][k0 + 1];
        b[0] = W[(size_t)k0 * NHID + n0 + mn];
        b[1] = W[(size_t)(k0 + 1) * NHID + n0 + mn];
        acc = __builtin_amdgcn_wmma_f32_16x16x4_f32(false, a, false, b,
                                                    (short)0, acc, false, false);
    }

    float bv = bias[n0 + mn];
    #pragma unroll
    for (int v = 0; v < 8; ++v) {
        int m = v + 8 * half;
        float z = acc[v] + bv;
        hout[(size_t)(tile * 16 + m) * NHID + (n0 + mn)] = z > 0.f ? z : 0.f;
    }
}

// -------------------------------- logits = [x1|x2] @ Wfc + bfc   (K=256, Ncols=40)
// 3 waves per block: n-tiles {0,16,32}; columns >= 40 masked on B-load and store
__global__ __launch_bounds__(96)
void gin_jkfc_kernel(const float* __restrict__ x1,
                     const float* __restrict__ x2,
                     const float* __restrict__ Wfc,
                     const float* __restrict__ bfc,
                     float* __restrict__ logits) {
    __shared__ float Xs[16][2 * NHID + 4];   // stride 260 -> conflict-free
    int tile = blockIdx.x;
    int tid  = threadIdx.x;

    for (int i = tid; i < 16 * 2 * NHID; i += 96) {
        int r = i >> 8;
        int c = i & (2 * NHID - 1);
        Xs[r][c] = (c < NHID) ? x1[(size_t)(tile * 16 + r) * NHID + c]
                              : x2[(size_t)(tile * 16 + r) * NHID + (c - NHID)];
    }
    __syncthreads();

    int wave = tid >> 5;
    int lane = tid & 31;
    int half = lane >> 4;
    int mn   = lane & 15;
    int n    = wave * 16 + mn;
    bool valid = (n < NCLASS);

    v8f acc = {0.f, 0.f, 0.f, 0.f, 0.f, 0.f, 0.f, 0.f};
    for (int kk = 0; kk < 2 * NHID; kk += 4) {
        int k0 = kk + 2 * half;
        v2f a, b;
        a[0] = Xs[mn][k0];
        a[1] = Xs[mn][k0 + 1];
        b[0] = valid ? Wfc[(size_t)k0 * NCLASS + n] : 0.f;
        b[1] = valid ? Wfc[(size_t)(k0 + 1) * NCLASS + n] : 0.f;
        acc = __builtin_amdgcn_wmma_f32_16x16x4_f32(false, a, false, b,
                                                    (short)0, acc, false, false);
    }

    if (valid) {
        float bv = bfc[n];
        #pragma unroll
        for (int v = 0; v < 8; ++v) {
            int m = v + 8 * half;
            logits[(size_t)(tile * 16 + m) * NCLASS + n] = acc[v] + bv;
        }
    }
}

// ---------------------------------------------------------- row-wise log_softmax
__global__ void gin_logsoftmax_kernel(const float* __restrict__ logits,
                                      float* __restrict__ out) {
    int node = blockIdx.x * blockDim.x + threadIdx.x;
    if (node >= N_NODES) return;
    const float* l = logits + (size_t)node * NCLASS;
    float m = l[0];
    #pragma unroll
    for (int c = 1; c < NCLASS; ++c) m = fmaxf(m, l[c]);
    float s = 0.f;
    #pragma unroll
    for (int c = 0; c < NCLASS; ++c) s += __expf(l[c] - m);
    float ls = __logf(s);
    float* o = out + (size_t)node * NCLASS;
    #pragma unroll
    for (int c = 0; c < NCLASS; ++c) o[c] = l[c] - m - ls;
}

extern "C" void kernel_launch(void* const* d_in, const int* in_sizes, int n_in,
                              void* d_out, int out_size, void* d_ws, size_t ws_size,
                              hipStream_t stream) {
    const float* x   = (const float*)d_in[0];
    const int*   ei  = (const int*)  d_in[1];   // [2, E] int32
    const float* ew  = (const float*)d_in[2];
    const float* W1  = (const float*)d_in[3];
    const float* b1  = (const float*)d_in[4];
    const float* W2  = (const float*)d_in[5];
    const float* b2  = (const float*)d_in[6];
    const float* Wfc = (const float*)d_in[7];
    const float* bfc = (const float*)d_in[8];
    float* out = (float*)d_out;

    const int* src = ei;
    const int* dst = ei + N_EDGES;

    float* agg    = (float*)d_ws;                       // N*128
    float* x1     = agg + (size_t)N_NODES * NHID;       // N*128
    float* x2     = x1  + (size_t)N_NODES * NHID;       // N*128
    float* logits = agg;                                // reuse agg after layer 2

    const int nfeat4      = N_NODES * NHID / 4;
    const int zero_blocks = (nfeat4 + 255) / 256;
    const int scat_blocks = (N_EDGES + 7) / 8;          // 8 waves (edges) per block
    const int tile_blocks = N_NODES / 16;               // 100000 % 16 == 0

    // layer 1
    gin_zero_kernel<<<zero_blocks, 256, 0, stream>>>(agg, nfeat4);
    gin_scatter_kernel<<<scat_blocks, 256, 0, stream>>>(x, src, dst, ew, agg);
    gin_gemm_relu_kernel<<<tile_blocks, 256, 0, stream>>>(x, agg, W1, b1, x1);
    // layer 2
    gin_zero_kernel<<<zero_blocks, 256, 0, stream>>>(agg, nfeat4);
    gin_scatter_kernel<<<scat_blocks, 256, 0, stream>>>(x1, src, dst, ew, agg);
    gin_gemm_relu_kernel<<<tile_blocks, 256, 0, stream>>>(x1, agg, W2, b2, x2);
    // JK concat + FC + log_softmax
    gin_jkfc_kernel<<<tile_blocks, 96, 0, stream>>>(x1, x2, Wfc, bfc, logits);
    gin_logsoftmax_kernel<<<(N_NODES + 255) / 256, 256, 0, stream>>>(logits, out);
}